// DynamicFeatureSelection_25666724561022
// MI455X (gfx1250) — compile-verified
//
#include <hip/hip_runtime.h>

typedef float v2f __attribute__((ext_vector_type(2)));
typedef float v8f __attribute__((ext_vector_type(8)));

#define BATCH 4
#define CTOT  3840
#define HW    1024
#define DDIM  4

// One wave per tile: 4 consecutive channels x 16 consecutive pixels, one batch.
// out[b,d,c,p] = relu(w[seg(c),d] * feat[b,c,p]) via V_WMMA_F32_16X16X4_F32:
//   D(16x16) = A(16x4) * B(4x16),  A[m,k] = (k == m>>2) ? w[seg, m&3] : 0,
//   B[k,n]   = feat[c_base + k, pixel n].
__global__ __launch_bounds__(256) void dfs_wmma_kernel(
    const float* __restrict__ f0, const float* __restrict__ f1,
    const float* __restrict__ f2, const float* __restrict__ f3,
    const float* __restrict__ w,  float* __restrict__ out)
{
    const int lane = threadIdx.x & 31;
    const int hf   = lane >> 4;   // half-wave 0/1
    const int nl   = lane & 15;   // N (pixel) index for B/D, M index for A

    // wave tile id: fastest dim = pixel tile (keeps inter-wave stores contiguous)
    const int wave = (blockIdx.x << 3) | (threadIdx.x >> 5);
    const int pt = wave & 63;          // 64 pixel tiles of 16 per (b,c) plane
    const int t  = wave >> 6;
    const int cb = t % 960;            // channel quad
    const int b  = t / 960;            // batch
    const int c_base = cb << 2;

    // segment select — wave-uniform, quad never straddles a boundary
    const int seg = (c_base >= 256) + (c_base >= 768) + (c_base >= 1792);
    const float* fp  = (seg == 0) ? f0  : (seg == 1) ? f1  : (seg == 2) ? f2   : f3;
    const int Cseg   = (seg == 0) ? 256 : (seg == 1) ? 512 : (seg == 2) ? 1024 : 2048;
    const int segoff = (seg == 0) ? 0   : (seg == 1) ? 256 : (seg == 2) ? 768  : 1792;
    const int lc = c_base - segoff;

    const int px = (pt << 4) + nl;

    // ---- B (4x16, KxN): lane nl = pixel; VGPR0 K=2*hf, VGPR1 K=2*hf+1.
    // Each load: 16 consecutive floats per half-wave -> coalesced 64B runs.
    const int k0 = hf << 1;
    v2f B;
    B.x = fp[(b * Cseg + lc + k0)     * HW + px];
    B.y = fp[(b * Cseg + lc + k0 + 1) * HW + px];

    // ---- A (16x4, MxK): lane's M = nl; m = 4*c_local + d.
    const int d_a  = nl & 3;
    const int cl_a = nl >> 2;
    const float wv = w[(seg << 2) + d_a];   // tiny, cache-resident
    v2f A;
    A.x = (cl_a == k0)     ? wv : 0.0f;
    A.y = (cl_a == k0 + 1) ? wv : 0.0f;

    v8f C = {};
    C = __builtin_amdgcn_wmma_f32_16x16x4_f32(
            /*neg_a=*/false, A, /*neg_b=*/false, B,
            /*c_mod=*/(short)0, C, /*reuse_a=*/false, /*reuse_b=*/false);

    // ---- relu + store. D VGPR r: lanes0-15 M=r, lanes16-31 M=r+8.
    // Each store_b32 covers two contiguous 64B row segments -> coalesced.
#pragma unroll
    for (int r = 0; r < 8; ++r) {
        const int M  = r + (hf << 3);
        const int d  = M & 3;
        const int cl = M >> 2;
        const int o  = ((b * DDIM + d) * CTOT + c_base + cl) * HW + px;
        out[o] = fmaxf(C[r], 0.0f);
    }
}

extern "C" void kernel_launch(void* const* d_in, const int* in_sizes, int n_in,
                              void* d_out, int out_size, void* d_ws, size_t ws_size,
                              hipStream_t stream) {
    const float* f0 = (const float*)d_in[0];
    const float* f1 = (const float*)d_in[1];
    const float* f2 = (const float*)d_in[2];
    const float* f3 = (const float*)d_in[3];
    const float* w  = (const float*)d_in[4];
    float* out = (float*)d_out;
    // waves = B * (C/4) * (HW/16) = 4 * 960 * 64 = 245760; 8 wave32 per block
    dim3 grid(245760 / 8), block(256);
    hipLaunchKernelGGL(dfs_wmma_kernel, grid, block, 0, stream,
                       f0, f1, f2, f3, w, out);
}